// Xw4d_45543833206971
// MI455X (gfx1250) — compile-verified
//
#include <hip/hip_runtime.h>
#include <hip/hip_bf16.h>

#define NN 4096
#define KP1 4                       // k+1 nearest
#define COLS_PER_THREAD 4           // float4 per thread in push kernel
#define PUSH_STRIPS (NN / (256 * COLS_PER_THREAD))  // 4 column strips of 1024
#define PUSH_JROWS  32              // rows per block
#define PUSH_JBLKS  (NN / PUSH_JROWS)               // 128
#define PUSH_BLOCKS (PUSH_STRIPS * PUSH_JBLKS)      // 512

typedef __attribute__((ext_vector_type(2))) float v2f;
typedef __attribute__((ext_vector_type(8))) float v8f;

// ---------------------------------------------------------------------------
// Wave32 sum reduction via V_WMMA_F32_16X16X4_F32.
// A(16x4): VGPR0 = lane value (K=0 for lanes 0-15, K=2 for lanes 16-31),
//          VGPR1 = 0.  B = all ones (layout-independent).
// D[m][n] = acc[lane m] + acc[lane m+16]  (row sums, replicated over n).
// Lane l<16 holds D[r][l] in c[r] (M=0..7); lane>=16 holds M=8..15.
// Sum c[0..7] in-lane, then one xor-16 shuffle-add => full wave sum.
// EXEC must be all ones at the call site (no divergence).
// ---------------------------------------------------------------------------
__device__ __forceinline__ float wave_reduce_wmma(float x) {
    v2f a; a[0] = x;    a[1] = 0.0f;
    v2f b; b[0] = 1.0f; b[1] = 1.0f;
    v8f c = {};
    c = __builtin_amdgcn_wmma_f32_16x16x4_f32(
            /*neg_a=*/false, a, /*neg_b=*/false, b,
            /*c_mod=*/(short)0, c, /*reuse_a=*/false, /*reuse_b=*/false);
    float s = c[0] + c[1] + c[2] + c[3] + c[4] + c[5] + c[6] + c[7];
    s += __shfl_xor(s, 16, 32);   // lanes 0-15 hold rows 0-7 sum, 16-31 rows 8-15
    return s;                     // all lanes: wave total
}

// Stable argsort ordering: smaller value first, ties broken by smaller index.
__device__ __forceinline__ bool kvLess(float av, int ai, float bv, int bi) {
    return (av < bv) || (av == bv && ai < bi);
}

// ---------------------------------------------------------------------------
// Kernel 1: per-row 4 smallest (value,index); apply same-class/non-self mask;
// write thresholds thr[i*4+m] = mask ? val+1 : -inf, and pull_partial[i].
// One 256-thread block per row; b128 coalesced scan + LDS merge tree.
// ---------------------------------------------------------------------------
__global__ __launch_bounds__(256) void lmnn_top4_kernel(
        const float* __restrict__ D, const int* __restrict__ labels,
        float* __restrict__ thr, float* __restrict__ pull_partial) {
    __shared__ float sv[256 * KP1];
    __shared__ int   si[256 * KP1];

    const int row = blockIdx.x;
    const int tid = threadIdx.x;
    const float4* __restrict__ drow4 =
        reinterpret_cast<const float4*>(D + (size_t)row * NN);

    float v[KP1]  = { __builtin_inff(), __builtin_inff(),
                      __builtin_inff(), __builtin_inff() };
    int   id[KP1] = { 0x7FFFFFFF, 0x7FFFFFFF, 0x7FFFFFFF, 0x7FFFFFFF };

    // 4 x float4 per thread = 16 elements, global_load_b128 coalesced.
    #pragma unroll
    for (int t = 0; t < NN / (256 * 4); ++t) {
        const int q = t * 256 + tid;            // float4 index within row
        const float4 x4 = drow4[q];
        const float xe[4] = { x4.x, x4.y, x4.z, x4.w };
        #pragma unroll
        for (int e = 0; e < 4; ++e) {
            const int   col = q * 4 + e;
            const float x   = xe[e];
            if (kvLess(x, col, v[3], id[3])) {
                v[3] = x; id[3] = col;
                #pragma unroll
                for (int m = 3; m > 0; --m) {   // compare-swap bubble
                    if (kvLess(v[m], id[m], v[m - 1], id[m - 1])) {
                        float tv = v[m]; v[m] = v[m - 1]; v[m - 1] = tv;
                        int   ti = id[m]; id[m] = id[m - 1]; id[m - 1] = ti;
                    }
                }
            }
        }
    }

    #pragma unroll
    for (int m = 0; m < KP1; ++m) { sv[tid * KP1 + m] = v[m]; si[tid * KP1 + m] = id[m]; }
    __syncthreads();

    // Deterministic merge tree: 256 sorted 4-lists -> 1 sorted 4-list.
    for (int s = 128; s > 0; s >>= 1) {
        if (tid < s) {
            float a[KP1], bl[KP1]; int ai[KP1], bi[KP1];
            #pragma unroll
            for (int m = 0; m < KP1; ++m) {
                a[m]  = sv[tid * KP1 + m];        ai[m] = si[tid * KP1 + m];
                bl[m] = sv[(tid + s) * KP1 + m];  bi[m] = si[(tid + s) * KP1 + m];
            }
            float o[KP1]; int oi[KP1];
            int pa = 0, pb = 0;
            #pragma unroll
            for (int r = 0; r < KP1; ++r) {       // 2-pointer merge, take 4 smallest
                const bool ta = kvLess(a[pa], ai[pa], bl[pb], bi[pb]);
                o[r]  = ta ? a[pa]  : bl[pb];
                oi[r] = ta ? ai[pa] : bi[pb];
                pa += ta ? 1 : 0; pb += ta ? 0 : 1;
            }
            #pragma unroll
            for (int m = 0; m < KP1; ++m) { sv[tid * KP1 + m] = o[m]; si[tid * KP1 + m] = oi[m]; }
        }
        __syncthreads();
    }

    if (tid == 0) {
        const int li = labels[row];
        float pull = 0.0f;
        #pragma unroll
        for (int m = 0; m < KP1; ++m) {
            const float val = sv[m];
            const int   j   = si[m];
            const bool mask = (j != row) && (labels[j] == li);
            pull += mask ? val : 0.0f;
            // -inf threshold => relu(t - d) == 0, folds the pull_mask away.
            thr[row * KP1 + m] = mask ? (val + 1.0f) : -__builtin_inff();
        }
        pull_partial[row] = pull;
    }
}

// ---------------------------------------------------------------------------
// Kernel 2: push = sum_{i,m,j} diff(i,j) * relu(thr[i,m] - D[j,i]).
// Each thread owns 4 adjacent columns (thresholds + labels in registers),
// streams D row-major with global_load_b128 (L2-resident after kernel 1).
// WMMA wave reduction, deterministic per-block partial.
// ---------------------------------------------------------------------------
__global__ __launch_bounds__(256) void lmnn_push_kernel(
        const float* __restrict__ D, const int* __restrict__ labels,
        const float* __restrict__ thr, float* __restrict__ push_partial) {
    const int tid   = threadIdx.x;
    const int strip = blockIdx.x & (PUSH_STRIPS - 1);
    const int jblk  = blockIdx.x / PUSH_STRIPS;
    const int i0    = strip * (256 * COLS_PER_THREAD) + tid * COLS_PER_THREAD;

    const float4* __restrict__ thr4 = reinterpret_cast<const float4*>(thr);
    const float4 t0 = thr4[i0 + 0];
    const float4 t1 = thr4[i0 + 1];
    const float4 t2 = thr4[i0 + 2];
    const float4 t3 = thr4[i0 + 3];
    const int4 lab4 = reinterpret_cast<const int4*>(labels)[i0 / 4];

    float acc = 0.0f;
    const int j0 = jblk * PUSH_JROWS;
    #pragma unroll 4
    for (int j = j0; j < j0 + PUSH_JROWS; ++j) {
        const int lj = labels[j];                                 // uniform
        const float4 d4 =
            reinterpret_cast<const float4*>(D + (size_t)j * NN)[i0 / 4];
        const float r0 = fmaxf(t0.x - d4.x, 0.0f) + fmaxf(t0.y - d4.x, 0.0f)
                       + fmaxf(t0.z - d4.x, 0.0f) + fmaxf(t0.w - d4.x, 0.0f);
        const float r1 = fmaxf(t1.x - d4.y, 0.0f) + fmaxf(t1.y - d4.y, 0.0f)
                       + fmaxf(t1.z - d4.y, 0.0f) + fmaxf(t1.w - d4.y, 0.0f);
        const float r2 = fmaxf(t2.x - d4.z, 0.0f) + fmaxf(t2.y - d4.z, 0.0f)
                       + fmaxf(t2.z - d4.z, 0.0f) + fmaxf(t2.w - d4.z, 0.0f);
        const float r3 = fmaxf(t3.x - d4.w, 0.0f) + fmaxf(t3.y - d4.w, 0.0f)
                       + fmaxf(t3.z - d4.w, 0.0f) + fmaxf(t3.w - d4.w, 0.0f);
        acc += (lj != lab4.x) ? r0 : 0.0f;
        acc += (lj != lab4.y) ? r1 : 0.0f;
        acc += (lj != lab4.z) ? r2 : 0.0f;
        acc += (lj != lab4.w) ? r3 : 0.0f;
    }

    const float wsum = wave_reduce_wmma(acc);       // EXEC all-ones here

    __shared__ float swave[8];
    const int lane = tid & 31, wv = tid >> 5;
    if (lane == 0) swave[wv] = wsum;
    __syncthreads();
    if (tid == 0) {
        float s = 0.0f;
        #pragma unroll
        for (int w = 0; w < 8; ++w) s += swave[w];  // fixed order: deterministic
        push_partial[blockIdx.x] = s;
    }
}

// ---------------------------------------------------------------------------
// Kernel 3: deterministic fixed-order final reduction; writes scalar loss.
// loss = (1-mu)*pull + mu*push, mu = 0.5
// ---------------------------------------------------------------------------
__global__ __launch_bounds__(256) void lmnn_final_kernel(
        const float* __restrict__ pull_partial,
        const float* __restrict__ push_partial,
        float* __restrict__ out) {
    __shared__ float sp[256], sq[256];
    const int tid = threadIdx.x;
    float p = 0.0f, q = 0.0f;
    for (int t = tid; t < NN; t += 256)          p += pull_partial[t];
    for (int t = tid; t < PUSH_BLOCKS; t += 256) q += push_partial[t];
    sp[tid] = p; sq[tid] = q;
    __syncthreads();
    for (int s = 128; s > 0; s >>= 1) {
        if (tid < s) { sp[tid] += sp[tid + s]; sq[tid] += sq[tid + s]; }
        __syncthreads();
    }
    if (tid == 0) out[0] = 0.5f * sp[0] + 0.5f * sq[0];
}

extern "C" void kernel_launch(void* const* d_in, const int* in_sizes, int n_in,
                              void* d_out, int out_size, void* d_ws, size_t ws_size,
                              hipStream_t stream) {
    const float* D      = (const float*)d_in[0];   // [N*N] float32
    const int*   labels = (const int*)d_in[1];     // [N] int32
    float*       out    = (float*)d_out;           // scalar float32

    float* thr          = (float*)d_ws;            // [4*N]
    float* pull_partial = thr + (size_t)KP1 * NN;  // [N]
    float* push_partial = pull_partial + NN;       // [PUSH_BLOCKS]

    lmnn_top4_kernel <<<NN,          256, 0, stream>>>(D, labels, thr, pull_partial);
    lmnn_push_kernel <<<PUSH_BLOCKS, 256, 0, stream>>>(D, labels, thr, push_partial);
    lmnn_final_kernel<<<1,           256, 0, stream>>>(pull_partial, push_partial, out);
}